// DynamicGraphConvolution_21242908246180
// MI455X (gfx1250) — compile-verified
//
#include <hip/hip_runtime.h>
#include <hip/hip_bf16.h>

typedef __attribute__((ext_vector_type(2))) float v2f;
typedef __attribute__((ext_vector_type(4))) float v4f;
typedef __attribute__((ext_vector_type(8))) float v8f;

#define B_SZ   16
#define N_SZ   4096
#define F_SZ   64
#define K_SZ   16
#define NTILE  (N_SZ / 16)          // 256 column tiles
#define NINF   (-3.402823466e38f)

// ---------------------------------------------------------------------------
// Kernel 1: support = X @ W (WMMA f32 16x16x4), plus xx[n] = ||x_n||^2
// grid: B*N/16 blocks of 32 threads (one wave per 16-row tile)
// ---------------------------------------------------------------------------
__global__ __launch_bounds__(32)
void support_xx_kernel(const float* __restrict__ X, const float* __restrict__ W,
                       float* __restrict__ support, float* __restrict__ xx)
{
    const int lane = threadIdx.x;
    const int tile = blockIdx.x;
    const int b    = tile >> 8;          // N/16 = 256 tiles per batch
    const int i0   = (tile & 255) << 4;
    const int m    = lane & 15;          // A-row / B-col / D-col
    const int h    = lane >> 4;          // K-half selector

    const float* xrow = X + ((size_t)(b * N_SZ + i0 + m)) * F_SZ + 2 * h;

    v2f a[16];
    float ss = 0.0f;
#pragma unroll
    for (int c = 0; c < 16; ++c) {
        a[c] = *(const v2f*)(xrow + 4 * c);
        ss += a[c].x * a[c].x + a[c].y * a[c].y;
    }
    ss += __shfl_xor(ss, 16, 32);
    if (h == 0) xx[b * N_SZ + i0 + m] = ss;

#pragma unroll
    for (int n0t = 0; n0t < 4; ++n0t) {
        const int n0 = n0t * 16;
        v8f acc = {};
#pragma unroll
        for (int c = 0; c < 16; ++c) {
            const int k = 4 * c + 2 * h;
            v2f bf;
            bf.x = W[(k + 0) * F_SZ + n0 + m];
            bf.y = W[(k + 1) * F_SZ + n0 + m];
            acc = __builtin_amdgcn_wmma_f32_16x16x4_f32(
                false, a[c], false, bf, (short)0, acc, false, false);
        }
#pragma unroll
        for (int r = 0; r < 8; ++r) {
            support[((size_t)(b * N_SZ + i0 + r + 8 * h)) * F_SZ + n0 + m] = acc[r];
        }
    }
}

// insert into a register-resident descending top-16 (strict compares = stable)
#define TOPK_INSERT(v, id)                                                   \
    do {                                                                     \
        if ((v) > vals[15]) {                                                \
            vals[15] = (v); idxs[15] = (id);                                 \
            _Pragma("unroll")                                                \
            for (int s = 15; s > 0; --s) {                                   \
                if (vals[s] > vals[s - 1]) {                                 \
                    float tv = vals[s]; vals[s] = vals[s - 1]; vals[s - 1] = tv; \
                    int   ti = idxs[s]; idxs[s] = idxs[s - 1]; idxs[s - 1] = ti; \
                }                                                            \
            }                                                                \
        }                                                                    \
    } while (0)

// ---------------------------------------------------------------------------
// Kernel 2: fused pairwise (WMMA f32, async-LDS double-buffered B-tile) ->
//           streaming per-lane top-16 -> neighbor gather/mean + bias.
// grid: B*N/128 blocks of 128 threads; each wave owns 32 rows.
// ---------------------------------------------------------------------------
__global__ __launch_bounds__(128)
void knn_gather_kernel(const float* __restrict__ X, const float* __restrict__ support,
                       const float* __restrict__ xx, const float* __restrict__ bias,
                       float* __restrict__ out)
{
    __shared__ __align__(16) float btile[2][16 * F_SZ];  // double-buffered B tile (2 x 4KB)
    __shared__ __align__(16) float tileD[4][32 * 20];    // per-wave D transpose (stride 20)
    __shared__ int             sel[4][32 * 16];          // per-wave selected indices

    const int tid  = threadIdx.x;
    const int lane = tid & 31;
    const int w    = tid >> 5;
    const int blk  = blockIdx.x;
    const int b    = blk >> 5;                 // N/128 = 32 blocks per batch
    const int i0w  = ((blk & 31) << 7) + (w << 5);   // this wave's 32-row base
    const int m    = lane & 15;
    const int h    = lane >> 4;

    const float* Xb  = X  + (size_t)b * N_SZ * F_SZ;
    const float* xxb = xx + (size_t)b * N_SZ;

    // A fragments for two 16-row tiles (kept in VGPRs for all 256 iterations)
    v2f a0[16], a1[16];
    const float* ar0 = Xb + (size_t)(i0w + m) * F_SZ + 2 * h;
    const float* ar1 = ar0 + 16 * F_SZ;
#pragma unroll
    for (int c = 0; c < 16; ++c) {
        a0[c] = *(const v2f*)(ar0 + 4 * c);
        a1[c] = *(const v2f*)(ar1 + 4 * c);
    }

    // lane L ranks row (i0w + L) over all columns
    const float xr = xxb[i0w + lane];

    float vals[16]; int idxs[16];
#pragma unroll
    for (int t = 0; t < 16; ++t) { vals[t] = NINF; idxs[t] = 0; }

    const uint32_t ldsb0 = (uint32_t)(size_t)&btile[0][0];
    const uint32_t ldsb1 = (uint32_t)(size_t)&btile[1][0];
    const uint32_t co    = (uint32_t)tid * 16u;   // 128 threads x 16B x 2 = 4KB tile

    // prologue: async-copy tile j=0 into buffer 0 (rows of X are contiguous)
    asm volatile("global_load_async_to_lds_b128 %0, %1, %2 offset:0"
                 :: "v"(ldsb0 + co), "v"(co), "s"(Xb) : "memory");
    asm volatile("global_load_async_to_lds_b128 %0, %1, %2 offset:2048"
                 :: "v"(ldsb0 + co), "v"(co), "s"(Xb) : "memory");
    asm volatile("s_wait_asynccnt 0x0" ::: "memory");
    __syncthreads();

    for (int jt = 0; jt < NTILE; ++jt) {
        const int j = jt << 4;
        const float* bt = &btile[jt & 1][0];

        // issue async copy of the next tile into the other buffer
        if (jt + 1 < NTILE) {
            const uint32_t dst = ((jt & 1) ? ldsb0 : ldsb1) + co;
            const uint32_t ga  = (uint32_t)(jt + 1) * (16u * F_SZ * 4u) + co;
            asm volatile("global_load_async_to_lds_b128 %0, %1, %2 offset:0"
                         :: "v"(dst), "v"(ga), "s"(Xb) : "memory");
            asm volatile("global_load_async_to_lds_b128 %0, %1, %2 offset:2048"
                         :: "v"(dst), "v"(ga), "s"(Xb) : "memory");
        }

        // 32 WMMAs: every B fragment (from LDS) feeds both row tiles
        v8f acc0 = {}, acc1 = {};
#pragma unroll
        for (int c = 0; c < 16; ++c) {
            const v2f bf = *(const v2f*)(bt + m * F_SZ + 4 * c + 2 * h);
            acc0 = __builtin_amdgcn_wmma_f32_16x16x4_f32(
                false, a0[c], false, bf, (short)0, acc0, false, false);
            acc1 = __builtin_amdgcn_wmma_f32_16x16x4_f32(
                false, a1[c], false, bf, (short)0, acc1, false, false);
        }

        // scatter both D tiles to wave-private LDS (rows 0..31, stride 20)
        float* td = &tileD[w][0];
#pragma unroll
        for (int r = 0; r < 8; ++r) {
            td[(r + 8 * h) * 20 + m]        = acc0[r];
            td[(16 + r + 8 * h) * 20 + m]   = acc1[r];
        }

        // transpose read: lane L gets row L's 16 distances (16B-aligned b128s)
        const float* trow = td + lane * 20;
        v4f d[4], xc[4];
#pragma unroll
        for (int q = 0; q < 4; ++q) {
            d[q]  = *(const v4f*)(trow + 4 * q);
            xc[q] = *(const v4f*)(xxb + j + 4 * q);
        }
#pragma unroll
        for (int q = 0; q < 4; ++q) {
#pragma unroll
            for (int c = 0; c < 4; ++c) {
                const float v = -((xr + xc[q][c]) - 2.0f * d[q][c]);
                TOPK_INSERT(v, j + 4 * q + c);
            }
        }

        asm volatile("s_wait_asynccnt 0x0" ::: "memory");
        __syncthreads();
    }

    // publish per-row index lists (wave-private region; in-wave DS order)
    int* sw = &sel[w][0];
#pragma unroll
    for (int t = 0; t < 16; ++t) sw[lane * 16 + t] = idxs[t];

    // gather 16 neighbors per row, mean + bias; 2 features per lane
    const float* supb = support + (size_t)b * N_SZ * F_SZ;
    const v2f bias2 = *(const v2f*)(bias + 2 * lane);
    for (int r = 0; r < 32; ++r) {
        v2f acc2; acc2.x = 0.0f; acc2.y = 0.0f;
#pragma unroll
        for (int t = 0; t < 16; ++t) {
            const int id = sw[r * 16 + t];
            const v2f s = *((const v2f*)(supb + (size_t)id * F_SZ) + lane);
            acc2.x += s.x; acc2.y += s.y;
        }
        v2f o;
        o.x = acc2.x * 0.0625f + bias2.x;
        o.y = acc2.y * 0.0625f + bias2.y;
        *((v2f*)(out + (size_t)(b * N_SZ + i0w + r) * F_SZ) + lane) = o;
    }
}

extern "C" void kernel_launch(void* const* d_in, const int* in_sizes, int n_in,
                              void* d_out, int out_size, void* d_ws, size_t ws_size,
                              hipStream_t stream) {
    const float* X    = (const float*)d_in[0];   // [B,N,64]
    const float* W    = (const float*)d_in[1];   // [64,64]
    const float* bias = (const float*)d_in[2];   // [64]
    // d_in[3] = k (hardcoded 16)
    float* out = (float*)d_out;

    float* xx      = (float*)d_ws;               // B*N floats   (256 KB)
    float* support = (float*)d_ws + B_SZ * N_SZ; // B*N*64 floats (16 MB)

    support_xx_kernel<<<B_SZ * (N_SZ / 16), 32, 0, stream>>>(X, W, support, xx);
    knn_gather_kernel<<<B_SZ * (N_SZ / 128), 128, 0, stream>>>(X, support, xx, bias, out);
}